// ChamferLoss_30966714204483
// MI455X (gfx1250) — compile-verified
//
#include <hip/hip_runtime.h>

typedef __attribute__((ext_vector_type(2))) float v2f;
typedef __attribute__((ext_vector_type(8))) float v8f;

// Exact branchless per-lane select: (lo & ~m) | (hi & m)  ->  v_bfi_b32.
// m is a per-lane constant (all-ones on lanes 16-31), so nothing can be
// sunk into an exec-masked region.
__device__ __forceinline__ float lane_sel(unsigned m, float hi, float lo) {
    return __uint_as_float((__float_as_uint(lo) & ~m) | (__float_as_uint(hi) & m));
}

// One wave handles a 16-row block of A; loops over all columns of B in 16-wide
// tiles. Each V_WMMA_F32_16X16X4_F32 produces the FULL squared-distance tile:
//   A (16x4): rows scaled by -2, K=3 column = 1.0
//   B (4x16): point coords,      K=3 row    = |b_col|^2
//   C       : per-row |a_row|^2  (C/D layout: VGPR k <-> row, constant per lane)
//   D = A*B + C = |a|^2 - 2 a.b + |b|^2
// Row-min tracked in squared space (sqrt is monotonic): one sqrt per row.
__global__ __launch_bounds__(128)
void chamfer_dir_kernel(const float* __restrict__ A, // [NA,3] rows: min over B
                        const float* __restrict__ B, // [NB,3]
                        int NA, int NB,
                        float* __restrict__ partial) // one float per wave
{
    const int lane      = threadIdx.x & 31;
    const int waveInBlk = threadIdx.x >> 5;
    const int waveId    = blockIdx.x * (blockDim.x >> 5) + waveInBlk;
    const int rowBase   = waveId * 16;
    if (rowBase >= NA) return;               // wave-uniform: EXEC stays all-1s

    const int half = lane >> 4;              // 0: K=0,1   1: K=2,3
    const int l16  = lane & 15;
    const unsigned hm = half ? 0xffffffffu : 0u;   // lane-constant select mask

    // ---- A tile (16x4 fp32), scaled by -2; K=3 column = 1.0 ----
    // ISA layout: lanes 0-15 VGPR{0,1} = K{0,1}; lanes 16-31 = K{2,3}; M = lane%16.
    const int arow = rowBase + l16;
    const float ax = A[arow * 3 + 0];
    const float ay = A[arow * 3 + 1];
    const float az = A[arow * 3 + 2];
    v2f a;
    a.x = lane_sel(hm, -2.0f * az, -2.0f * ax);
    a.y = lane_sel(hm, 1.0f,       -2.0f * ay);

    // ---- C operand: per-lane |a|^2 for the 8 rows this lane's C slots cover ----
    // C/D layout: VGPR k, lanes 0-15 -> M=k, lanes 16-31 -> M=8+k.
    v8f csqa;
#pragma unroll
    for (int k = 0; k < 8; ++k) {
        const int r = rowBase + half * 8 + k;
        const float x = A[r * 3 + 0], y = A[r * 3 + 1], z = A[r * 3 + 2];
        csqa[k] = x * x + y * y + z * z;
    }

    float runmin[8];                         // running min of squared distance
#pragma unroll
    for (int k = 0; k < 8; ++k) runmin[k] = 3.0e38f;

#pragma unroll 4
    for (int tile = 0; tile < NB; tile += 16) {
        const int col = tile + l16;
        const float b0 = B[col * 3 + 0];
        const float b1 = B[col * 3 + 1];
        const float b2 = B[col * 3 + 2];
        // computed unconditionally on every lane (keeps the b96 load merged,
        // no exec-masked sinking)
        const float sqb = fmaf(b0, b0, fmaf(b1, b1, b2 * b2));

        v2f b;                               // B tile 4x16; K=3 row carries |b|^2
        b.x = lane_sel(hm, b2,  b0);
        b.y = lane_sel(hm, sqb, b1);

        // D = (-2A|1) x (B|sqb) + sqa  ->  full squared-distance tile
        const v8f c = __builtin_amdgcn_wmma_f32_16x16x4_f32(
                /*neg_a=*/false, a, /*neg_b=*/false, b,
                /*c_mod=*/(short)0, csqa, /*reuse_a=*/false, /*reuse_b=*/false);

#pragma unroll
        for (int k = 0; k < 8; ++k)
            runmin[k] = fminf(runmin[k], c[k]);
    }

    // Min across the 16 lanes of each half-wave (masks 1..8 never cross halves).
#pragma unroll
    for (int k = 0; k < 8; ++k) {
        float m = runmin[k];
        m = fminf(m, __shfl_xor(m, 1, 32));
        m = fminf(m, __shfl_xor(m, 2, 32));
        m = fminf(m, __shfl_xor(m, 4, 32));
        m = fminf(m, __shfl_xor(m, 8, 32));
        // clamp + single sqrt per finished row-min (raw v_sqrt_f32, ~1 ulp)
        runmin[k] = __builtin_amdgcn_sqrtf(fmaxf(m, 0.0f));
    }

    // Each half-wave now holds 8 finished row distances (replicated over lanes).
    float local = 0.0f;
#pragma unroll
    for (int k = 0; k < 8; ++k) local += runmin[k];
    local += __shfl_xor(local, 16, 32);      // combine rows 0-7 with rows 8-15

    if (lane == 0) partial[waveId] = local;
}

// Deterministic fixed-order tree reduction of all per-wave partials.
__global__ __launch_bounds__(256)
void chamfer_reduce_kernel(const float* __restrict__ partial, int n,
                           float* __restrict__ out)
{
    __shared__ float smem[256];
    float s = 0.0f;
    for (int i = threadIdx.x; i < n; i += 256) s += partial[i];
    smem[threadIdx.x] = s;
    __syncthreads();
    for (int stride = 128; stride > 0; stride >>= 1) {
        if ((int)threadIdx.x < stride) smem[threadIdx.x] += smem[threadIdx.x + stride];
        __syncthreads();
    }
    if (threadIdx.x == 0) out[0] = smem[0] * 0.001f;
}

extern "C" void kernel_launch(void* const* d_in, const int* in_sizes, int n_in,
                              void* d_out, int out_size, void* d_ws, size_t ws_size,
                              hipStream_t stream) {
    (void)n_in; (void)out_size; (void)ws_size;
    const float* target = (const float*)d_in[0];   // [N,3]
    const float* output = (const float*)d_in[1];   // [M,3]
    const int N = in_sizes[0] / 3;
    const int M = in_sizes[1] / 3;

    float* partial = (float*)d_ws;
    const int w1 = M / 16;                         // waves: output -> target
    const int w2 = N / 16;                         // waves: target -> output

    const dim3 blk(128);                           // 4 waves / workgroup
    // loss_a_to_b: for each output point, min distance to target set
    chamfer_dir_kernel<<<dim3((w1 + 3) / 4), blk, 0, stream>>>(
        output, target, M, N, partial);
    // loss_b_to_a: for each target point, min distance to output set
    chamfer_dir_kernel<<<dim3((w2 + 3) / 4), blk, 0, stream>>>(
        target, output, N, M, partial + w1);

    chamfer_reduce_kernel<<<1, 256, 0, stream>>>(partial, w1 + w2, (float*)d_out);
}